// AdjModel_54872502174440
// MI455X (gfx1250) — compile-verified
//
#include <hip/hip_runtime.h>
#include <math.h>

#define B_   8
#define V_   5000
#define D_   10
#define EMBC 32
#define FC   2
#define EC   64
#define HC   4
#define DHC  16
#define M_   (B_*V_)     /* 40000 rows */
#define BIGF 1e9f
#define STEPF 0.01f
#define MOMF  0.9f

typedef float v2f __attribute__((ext_vector_type(2)));
typedef float v8f __attribute__((ext_vector_type(8)));
typedef unsigned int u32x4 __attribute__((ext_vector_type(4)));
typedef int i32x4 __attribute__((ext_vector_type(4)));
typedef int i32x8 __attribute__((ext_vector_type(8)));

__device__ __forceinline__ float sigmoidf_(float x) { return 1.f / (1.f + expf(-x)); }

// ---------------------------------------------------------------------------
// WMMA GEMM: C[M,N] = A[M,K] @ W[K,N] (+bias) (tanh).  fp32 WMMA 16x16x4.
// - Weight matrix W (K x N, <=48KB) is DMA'd to LDS once per workgroup by the
//   Tensor Data Mover (wave 0 issues tensor_load_to_lds, s_wait_tensorcnt,
//   then workgroup barrier).
// - Each wave register-blocks 4 M-tiles (64 rows) x one 16-col N-tile, so a
//   single B fragment (from LDS) feeds 4 consecutive WMMAs.
// Requires M%64==0, N%16==0, K%4==0.
// ---------------------------------------------------------------------------
template <int EPI> // bit0: add bias[col]; bit1: tanh
__global__ __launch_bounds__(256) void wmma_gemm(
    const float* __restrict__ A, const float* __restrict__ W,
    const float* __restrict__ bias, float* __restrict__ C,
    int M, int K, int N)
{
  extern __shared__ float lw[]; // K*N floats, filled by TDM

  // ---- stage W -> LDS via Tensor Data Mover (one descriptor, wave 0) ----
  if ((threadIdx.x >> 5) == 0) {
    unsigned long long ga = (unsigned long long)(const void*)W;
    u32x4 g0;
    g0[0] = 1u;                                    // count=1 (valid user D#)
    g0[1] = 0u;                                    // lds_addr = dynamic-LDS base
    g0[2] = (unsigned)(ga & 0xFFFFFFFFu);          // global_addr[31:0]
    g0[3] = (unsigned)((ga >> 32) & 0x1FFFFFFu)    // global_addr[56:32]
            | (2u << 30);                          // type=2 ("image")
    i32x8 g1;
    g1[0] = (int)(2u << 16);                       // data_size = 4 bytes
    g1[1] = (int)(((unsigned)N & 0xFFFFu) << 16);  // tensor_dim0[15:0]
    g1[2] = (int)(((unsigned)K & 0xFFFFu) << 16) | ((N >> 16) & 0xFFFF); // dim1 lo|dim0 hi
    g1[3] = (int)(((unsigned)N & 0xFFFFu) << 16);  // tile_dim0 = N | dim1 hi
    g1[4] = K & 0xFFFF;                            // tile_dim1 = K (tile_dim2=0)
    g1[5] = N;                                     // tensor_dim0_stride[31:0]
    g1[6] = 0;                                     // stride hi / dim1_stride lo
    g1[7] = 0;
    i32x4 gz; gz[0] = gz[1] = gz[2] = gz[3] = 0;
#if __clang_major__ >= 23
    i32x8 gz8;
    gz8[0]=gz8[1]=gz8[2]=gz8[3]=gz8[4]=gz8[5]=gz8[6]=gz8[7]=0;
    __builtin_amdgcn_tensor_load_to_lds(g0, g1, gz, gz, gz8, 0);
#else
    __builtin_amdgcn_tensor_load_to_lds(g0, g1, gz, gz, 0);
#endif
    __builtin_amdgcn_s_wait_tensorcnt(0);
  }
  __syncthreads();

  const int wpb    = blockDim.x >> 5;
  const int wave   = blockIdx.x * wpb + (threadIdx.x >> 5);
  const int lane   = threadIdx.x & 31;
  const int tilesN = N >> 4;
  const int tiles  = (M >> 6) * tilesN;  // 64 rows per wave
  if (wave >= tiles) return;
  const int m0  = (wave / tilesN) << 6;
  const int n0  = (wave % tilesN) << 4;
  const int l15 = lane & 15;
  const int hi  = lane >> 4;

  v8f acc0, acc1, acc2, acc3;
#pragma unroll
  for (int i = 0; i < 8; ++i) { acc0[i]=0.f; acc1[i]=0.f; acc2[i]=0.f; acc3[i]=0.f; }

  // A fragment pointers for the 4 M-tiles (ISA 7.12.2 32-bit A layout:
  // lanes0-15 rows, v0={K+0|K+2}, v1={K+1|K+3})
  const float* a0 = A + (size_t)(m0 +  0 + l15) * K + (hi << 1);
  const float* a1 = A + (size_t)(m0 + 16 + l15) * K + (hi << 1);
  const float* a2 = A + (size_t)(m0 + 32 + l15) * K + (hi << 1);
  const float* a3 = A + (size_t)(m0 + 48 + l15) * K + (hi << 1);
  const float* wl = lw + (size_t)(hi << 1) * N + n0 + l15;

  for (int kk = 0; kk < K; kk += 4) {
    v2f bf; bf.x = wl[0]; bf.y = wl[N];   // B fragment from LDS (shared by 4 WMMAs)
    v2f f0; f0.x = a0[0]; f0.y = a0[1];
    v2f f1; f1.x = a1[0]; f1.y = a1[1];
    v2f f2; f2.x = a2[0]; f2.y = a2[1];
    v2f f3; f3.x = a3[0]; f3.y = a3[1];
    acc0 = __builtin_amdgcn_wmma_f32_16x16x4_f32(false, f0, false, bf, (short)0, acc0, false, false);
    acc1 = __builtin_amdgcn_wmma_f32_16x16x4_f32(false, f1, false, bf, (short)0, acc1, false, false);
    acc2 = __builtin_amdgcn_wmma_f32_16x16x4_f32(false, f2, false, bf, (short)0, acc2, false, false);
    acc3 = __builtin_amdgcn_wmma_f32_16x16x4_f32(false, f3, false, bf, (short)0, acc3, false, false);
    a0 += 4; a1 += 4; a2 += 4; a3 += 4;
    wl += (size_t)4 * N;
  }

  const int col = n0 + l15;
  const float bb = (EPI & 1) ? bias[col] : 0.f;
#pragma unroll
  for (int j = 0; j < 8; ++j) {
    const int r = j + (hi << 3);
    float v0 = acc0[j] + bb, v1 = acc1[j] + bb, v2 = acc2[j] + bb, v3 = acc3[j] + bb;
    if (EPI & 2) { v0 = tanhf(v0); v1 = tanhf(v1); v2 = tanhf(v2); v3 = tanhf(v3); }
    C[(size_t)(m0 +  0 + r) * N + col] = v0;
    C[(size_t)(m0 + 16 + r) * N + col] = v1;
    C[(size_t)(m0 + 32 + r) * N + col] = v2;
    C[(size_t)(m0 + 48 + r) * N + col] = v3;
  }
}

// ---------------------------------------------------------------------------
__global__ void zero_accs_kernel(float* accs) {
  if (threadIdx.x < 32) accs[threadIdx.x] = 0.f;
}

// embedding rows clipped to max-norm 1:  emb * min(1, 1/max(||row||,1e-12))
__global__ void clip_emb_kernel(const float* __restrict__ ev, float* __restrict__ ec) {
  int vv = blockIdx.x * blockDim.x + threadIdx.x;
  if (vv >= V_) return;
  const float* r = ev + (size_t)vv * EMBC;
  float n2 = 0.f;
#pragma unroll
  for (int i = 0; i < EMBC; ++i) n2 += r[i] * r[i];
  float sc = fminf(1.f, 1.f / fmaxf(sqrtf(n2), 1e-12f));
  float* o = ec + (size_t)vv * EMBC;
#pragma unroll
  for (int i = 0; i < EMBC; ++i) o[i] = r[i] * sc;
}

// x = [emb | feat] @ W_enc + b_enc   (K=34 — scalar FMA loop, tiny)
__global__ void encoder_kernel(const float* __restrict__ ec, const float* __restrict__ feat,
                               const float* __restrict__ W, const float* __restrict__ b,
                               float* __restrict__ x) {
  int idx = blockIdx.x * blockDim.x + threadIdx.x;
  if (idx >= M_ * EC) return;
  int e = idx & (EC - 1);
  int bv = idx >> 6;
  int vv = bv % V_;
  float s = b[e];
  const float* er = ec + (size_t)vv * EMBC;
#pragma unroll
  for (int i = 0; i < EMBC; ++i) s += er[i] * W[i * EC + e];
  const float* fr = feat + (size_t)bv * FC;
  s += fr[0] * W[(EMBC + 0) * EC + e] + fr[1] * W[(EMBC + 1) * EC + e];
  x[idx] = s;
}

// masked multi-head attention over D=10 neighbors; thread per (b,v,h)
__global__ void attention_kernel(const float* __restrict__ q, const float* __restrict__ k,
                                 const float* __restrict__ v, const int* __restrict__ adj,
                                 const int* __restrict__ num_nodes, float* __restrict__ agg) {
  int idx = blockIdx.x * blockDim.x + threadIdx.x;
  if (idx >= M_ * HC) return;
  int h = idx % HC;
  int bv = idx / HC;
  int b = bv / V_;
  const int nn = num_nodes[b];
  const float* qrow = q + (size_t)bv * EC + h * DHC;
  float qr[DHC];
#pragma unroll
  for (int t = 0; t < DHC; ++t) qr[t] = qrow[t];
  float sc[D_];
#pragma unroll
  for (int d = 0; d < D_; ++d) {
    int nb = adj[(size_t)bv * D_ + d];
    if (nb == nn) { sc[d] = -BIGF; continue; }
    const float* kr = k + ((size_t)b * V_ + nb) * EC + h * DHC;
    float s = 0.f;
#pragma unroll
    for (int t = 0; t < DHC; ++t) s += qr[t] * kr[t];
    sc[d] = s * 0.25f; // 1/sqrt(16)
  }
  float mx = sc[0];
#pragma unroll
  for (int d = 1; d < D_; ++d) mx = fmaxf(mx, sc[d]);
  float sum = 0.f;
#pragma unroll
  for (int d = 0; d < D_; ++d) { sc[d] = expf(sc[d] - mx); sum += sc[d]; }
  const float inv = 1.f / sum;
  float a[DHC];
#pragma unroll
  for (int t = 0; t < DHC; ++t) a[t] = 0.f;
#pragma unroll
  for (int d = 0; d < D_; ++d) {
    int nb = adj[(size_t)bv * D_ + d];
    if (nb == nn) continue; // gathered vn is zero for pads
    const float* vr = v + ((size_t)b * V_ + nb) * EC + h * DHC;
    float w = sc[d] * inv;
#pragma unroll
    for (int t = 0; t < DHC; ++t) a[t] += w * vr[t];
  }
  float* o = agg + (size_t)bv * EC + h * DHC;
#pragma unroll
  for (int t = 0; t < DHC; ++t) o[t] = a[t];
}

// GRU combine: x = (1-z)*x + z*h   (gi already includes b_gru)
__global__ void gru_kernel(float* __restrict__ x, const float* __restrict__ gi,
                           const float* __restrict__ gh) {
  int idx = blockIdx.x * blockDim.x + threadIdx.x;
  if (idx >= M_ * EC) return;
  int e = idx & (EC - 1);
  int bv = idx >> 6;
  const float* gir = gi + (size_t)bv * (3 * EC);
  const float* ghr = gh + (size_t)bv * (3 * EC);
  float z = sigmoidf_(gir[e] + ghr[e]);
  float r = sigmoidf_(gir[EC + e] + ghr[EC + e]);
  float hh = tanhf(gir[2 * EC + e] + r * ghr[2 * EC + e]);
  float xv = x[idx];
  x[idx] = (1.f - z) * xv + z * hh;
}

// edge decoder: pred = tanh(S_self + T_neighbor + bd1) . Wd2 + bd2 - BIG*mask
__global__ void edge_pred_kernel(const float* __restrict__ S, const float* __restrict__ T,
                                 const float* __restrict__ bd1, const float* __restrict__ Wd2,
                                 const float* __restrict__ bd2, const int* __restrict__ adj,
                                 const int* __restrict__ num_nodes, float* __restrict__ pred) {
  int idx = blockIdx.x * blockDim.x + threadIdx.x;
  if (idx >= M_ * D_) return;
  int bv = idx / D_;
  int b = bv / V_;
  int nb = adj[idx];
  bool pad = (nb == num_nodes[b]);
  const float* Sr = S + (size_t)bv * 32;
  const float* Tr = T + ((size_t)b * V_ + (pad ? 0 : nb)) * 32;
  float w = 0.f;
#pragma unroll
  for (int i = 0; i < 32; ++i) {
    float pre = bd1[i] + (pad ? 0.f : (Sr[i] + Tr[i]));
    w += tanhf(pre) * Wd2[i];
  }
  pred[idx] = w + bd2[0] - (pad ? BIGF : 0.f);
}

// softmax over D per node + flow0 = normalized * max(demand,0)
__global__ void softmax_flowinit_kernel(const float* __restrict__ pred,
                                        const float* __restrict__ dem,
                                        float* __restrict__ nrm, float* __restrict__ flow) {
  int bv = blockIdx.x * blockDim.x + threadIdx.x;
  if (bv >= M_) return;
  float s[D_];
  const float* pr = pred + (size_t)bv * D_;
  float mx = pr[0];
#pragma unroll
  for (int d = 1; d < D_; ++d) mx = fmaxf(mx, pr[d]);
  float sum = 0.f;
#pragma unroll
  for (int d = 0; d < D_; ++d) { s[d] = expf(pr[d] - mx); sum += s[d]; }
  float inv = 1.f / sum;
  float dp = fmaxf(dem[bv], 0.f);
#pragma unroll
  for (int d = 0; d < D_; ++d) {
    float n = s[d] * inv;
    nrm[(size_t)bv * D_ + d] = n;
    flow[(size_t)bv * D_ + d] = n * dp;
  }
}

// one MCF iteration: dst = normalized * (gathered_in_sum + dem_pos)
__global__ void mcf_step_kernel(const float* __restrict__ src, const float* __restrict__ nrm,
                                const float* __restrict__ dem, const int* __restrict__ in_idx,
                                float* __restrict__ dst) {
  int bv = blockIdx.x * blockDim.x + threadIdx.x;
  if (bv >= M_) return;
  float tot = 0.f;
#pragma unroll
  for (int p = 0; p < D_; ++p) {
    const int* t = in_idx + ((size_t)bv * D_ + p) * 3;
    tot += src[((size_t)t[0] * V_ + t[1]) * D_ + t[2]];
  }
  float s = tot + fmaxf(dem[bv], 0.f);
#pragma unroll
  for (int d = 0; d < D_; ++d)
    dst[(size_t)bv * D_ + d] = nrm[(size_t)bv * D_ + d] * s;
}

// flow correction + per-batch sum of flow^2
__global__ void flow_cost_kernel(const float* __restrict__ flow, const int* __restrict__ rev,
                                 float* __restrict__ acc) {
  __shared__ float sd[256];
  int b = blockIdx.y;
  int i = blockIdx.x * blockDim.x + threadIdx.x;
  float val = 0.f;
  if (i < V_ * D_) {
    size_t idx = (size_t)b * V_ * D_ + i;
    const int* t = rev + idx * 3;
    float rv = flow[((size_t)t[0] * V_ + t[1]) * D_ + t[2]];
    float f = flow[idx];
    f = fmaxf(f - fminf(f, rv), 0.f);
    val = f * f;
  }
  sd[threadIdx.x] = val;
  __syncthreads();
  for (int s = 128; s > 0; s >>= 1) {
    if (threadIdx.x < s) sd[threadIdx.x] += sd[threadIdx.x + s];
    __syncthreads();
  }
  if (threadIdx.x == 0) atomicAdd(&acc[b], sd[0]);
}

// dv = uh . Wu2 + bu2;  also accumulate dual_demand = sum dv*demand per batch
__global__ void dv_kernel(const float* __restrict__ uh, const float* __restrict__ Wu2,
                          const float* __restrict__ bu2, const float* __restrict__ dem,
                          float* __restrict__ dv, float* __restrict__ ddem_acc) {
  __shared__ float sd[256];
  int b = blockIdx.y;
  int vv = blockIdx.x * blockDim.x + threadIdx.x;
  float contrib = 0.f;
  if (vv < V_) {
    size_t bv = (size_t)b * V_ + vv;
    float s = bu2[0];
    const float* ur = uh + bv * 32;
#pragma unroll
    for (int i = 0; i < 32; ++i) s += ur[i] * Wu2[i];
    dv[bv] = s;
    contrib = s * dem[bv];
  }
  sd[threadIdx.x] = contrib;
  __syncthreads();
  for (int s = 128; s > 0; s >>= 1) {
    if (threadIdx.x < s) sd[threadIdx.x] += sd[threadIdx.x + s];
    __syncthreads();
  }
  if (threadIdx.x == 0) atomicAdd(&ddem_acc[b], sd[0]);
}

// dual PGD (10 iters, fully in-register) + per-batch sum of df^2 + dd*df
__global__ void dual_pg_kernel(const float* __restrict__ dv, const int* __restrict__ adj,
                               const int* __restrict__ num_nodes, float* __restrict__ acc) {
  __shared__ float sd[256];
  int b = blockIdx.y;
  int i = blockIdx.x * blockDim.x + threadIdx.x;
  float contrib = 0.f;
  if (i < V_ * D_) {
    size_t idx = (size_t)b * V_ * D_ + i;
    int nb = adj[idx];
    bool pad = (nb == num_nodes[b]);
    if (!pad) {
      float dd = dv[(size_t)b * V_ + nb]; // dual_tr - dual (dual==0 when unmasked)
      float f = 0.f, a = 0.f;
#pragma unroll
      for (int it = 0; it < 10; ++it) {
        a = MOMF * a - STEPF * (2.f * f + dd);
        f = fmaxf(f + a, 0.f);
      }
      contrib = f * f + dd * f;
    } // pad: f forced to 0 every iter -> contrib 0
  }
  sd[threadIdx.x] = contrib;
  __syncthreads();
  for (int s = 128; s > 0; s >>= 1) {
    if (threadIdx.x < s) sd[threadIdx.x] += sd[threadIdx.x + s];
    __syncthreads();
  }
  if (threadIdx.x == 0) atomicAdd(&acc[b], sd[0]);
}

__global__ void final_kernel(const float* __restrict__ accs, float* __restrict__ out) {
  int b = threadIdx.x;
  if (b < B_) out[b] = accs[b] /*flow*/ - accs[8 + b] /*pg*/ + accs[16 + b] /*ddem*/;
}

// ---------------------------------------------------------------------------
extern "C" void kernel_launch(void* const* d_in, const int* in_sizes, int n_in,
                              void* d_out, int out_size, void* d_ws, size_t ws_size,
                              hipStream_t stream) {
  const float* demands = (const float*)d_in[0];
  const float* feat    = (const float*)d_in[1];
  const int*   adj     = (const int*)d_in[2];
  const int*   in_idx  = (const int*)d_in[3];
  const int*   rev_idx = (const int*)d_in[4];
  const int*   nn      = (const int*)d_in[5];
  const float* emb_var = (const float*)d_in[6];
  const float* W_enc   = (const float*)d_in[7];
  const float* b_enc   = (const float*)d_in[8];
  const float* Wq      = (const float*)d_in[9];
  const float* Wk      = (const float*)d_in[10];
  const float* Wv      = (const float*)d_in[11];
  const float* Wo      = (const float*)d_in[12];
  const float* W_gru   = (const float*)d_in[13];
  const float* U_gru   = (const float*)d_in[14];
  const float* b_gru   = (const float*)d_in[15];
  const float* Wd1     = (const float*)d_in[16];
  const float* bd1     = (const float*)d_in[17];
  const float* Wd2     = (const float*)d_in[18];
  const float* bd2     = (const float*)d_in[19];
  const float* Wu1     = (const float*)d_in[20];
  const float* bu1     = (const float*)d_in[21];
  const float* Wu2     = (const float*)d_in[22];
  const float* bu2     = (const float*)d_in[23];
  float* out = (float*)d_out;

  // workspace layout (floats); dead buffers are aliased
  float* ws   = (float*)d_ws;
  float* embc = ws;                       // 160,000
  float* x    = embc + 160000;            // 2,560,000
  float* q    = x + 2560000;              // q|k|v = 7,680,000 (aliased by gi)
  float* kb   = q + 2560000;
  float* vb   = kb + 2560000;
  float* gi   = q;                        // alias: q/k/v dead after attention
  float* agg  = vb + 2560000;             // 2,560,000
  float* nxt  = agg + 2560000;            // 2,560,000
  float* gh   = nxt + 2560000;            // 7,680,000
  float* S    = gh;                       // alias: gh dead after last GRU
  float* T    = gh + 1280000;
  float* uh   = gh + 2560000;
  float* dv   = agg;                      // alias: agg dead (40,000)
  float* accs = agg + 40000;              // 32 floats: flow[8] | pg[8] | ddem[8]
  float* pred = nxt;                      // alias: nxt dead (400,000 each)
  float* nrm  = nxt + 400000;
  float* flowA = nxt + 800000;
  float* flowB = nxt + 1200000;

  const int TB = 256;
  auto cdiv = [](int a, int b) { return (a + b - 1) / b; };
  // 4 M-tiles (64 rows) per wave, 8 waves per block
  auto gemmBlocks = [&](int M, int N) { return cdiv((M >> 6) * (N >> 4), 8); };
  auto gemmShmem  = [](int K, int N) { return (size_t)K * N * sizeof(float); };

  zero_accs_kernel<<<1, 64, 0, stream>>>(accs);
  clip_emb_kernel<<<cdiv(V_, TB), TB, 0, stream>>>(emb_var, embc);
  encoder_kernel<<<cdiv(M_ * EC, TB), TB, 0, stream>>>(embc, feat, W_enc, b_enc, x);

  for (int layer = 0; layer < 2; ++layer) {
    wmma_gemm<0><<<gemmBlocks(M_, EC), TB, gemmShmem(EC, EC), stream>>>(x, Wq, nullptr, q, M_, EC, EC);
    wmma_gemm<0><<<gemmBlocks(M_, EC), TB, gemmShmem(EC, EC), stream>>>(x, Wk, nullptr, kb, M_, EC, EC);
    wmma_gemm<0><<<gemmBlocks(M_, EC), TB, gemmShmem(EC, EC), stream>>>(x, Wv, nullptr, vb, M_, EC, EC);
    attention_kernel<<<cdiv(M_ * HC, TB), TB, 0, stream>>>(q, kb, vb, adj, nn, agg);
    wmma_gemm<2><<<gemmBlocks(M_, EC), TB, gemmShmem(EC, EC), stream>>>(agg, Wo, nullptr, nxt, M_, EC, EC);
    wmma_gemm<1><<<gemmBlocks(M_, 3 * EC), TB, gemmShmem(EC, 3 * EC), stream>>>(nxt, W_gru, b_gru, gi, M_, EC, 3 * EC);
    wmma_gemm<0><<<gemmBlocks(M_, 3 * EC), TB, gemmShmem(EC, 3 * EC), stream>>>(x, U_gru, nullptr, gh, M_, EC, 3 * EC);
    gru_kernel<<<cdiv(M_ * EC, TB), TB, 0, stream>>>(x, gi, gh);
  }

  // edge decoder split: S = x @ Wd1[0:64,:], T = x @ Wd1[64:128,:]
  wmma_gemm<0><<<gemmBlocks(M_, 32), TB, gemmShmem(EC, 32), stream>>>(x, Wd1, nullptr, S, M_, EC, 32);
  wmma_gemm<0><<<gemmBlocks(M_, 32), TB, gemmShmem(EC, 32), stream>>>(x, Wd1 + 64 * 32, nullptr, T, M_, EC, 32);
  // dual decoder hidden: uh = tanh(x @ Wu1 + bu1)
  wmma_gemm<3><<<gemmBlocks(M_, 32), TB, gemmShmem(EC, 32), stream>>>(x, Wu1, bu1, uh, M_, EC, 32);

  edge_pred_kernel<<<cdiv(M_ * D_, TB), TB, 0, stream>>>(S, T, bd1, Wd2, bd2, adj, nn, pred);
  softmax_flowinit_kernel<<<cdiv(M_, TB), TB, 0, stream>>>(pred, demands, nrm, flowA);

  for (int it = 0; it < 10; ++it) {
    const float* src = (it & 1) ? flowB : flowA;
    float* dst = (it & 1) ? flowA : flowB;
    mcf_step_kernel<<<cdiv(M_, TB), TB, 0, stream>>>(src, nrm, demands, in_idx, dst);
  }
  // 10 steps -> final flow in flowA

  dim3 gEdge(cdiv(V_ * D_, TB), B_);
  flow_cost_kernel<<<gEdge, TB, 0, stream>>>(flowA, rev_idx, accs);
  dim3 gNode(cdiv(V_, TB), B_);
  dv_kernel<<<gNode, TB, 0, stream>>>(uh, Wu2, bu2, demands, dv, accs + 16);
  dual_pg_kernel<<<gEdge, TB, 0, stream>>>(dv, adj, nn, accs + 8);
  final_kernel<<<1, 32, 0, stream>>>(accs, out);

  (void)in_sizes; (void)n_in; (void)out_size; (void)ws_size;
}